// TopKAttention_23553600651439
// MI455X (gfx1250) — compile-verified
//
#include <hip/hip_runtime.h>
#include <hip/hip_bf16.h>

// Problem constants (fixed by reference setup_inputs)
#define B_   8
#define L_   2048
#define D_   128
#define P_   64
#define TK_  64
#define ROWT 16                 // rows of scores per block
#define NCOLT (L_ / 16)         // 128 column tiles of 16
#define SLAB (ROWT * L_)        // 32768 scores per block (128 KB f32 in LDS)
#define NBLK_ROW (L_ / ROWT)    // 128 score blocks per batch
#define NCAND (NBLK_ROW * TK_)  // 8192 candidates per batch

typedef float v2f  __attribute__((ext_vector_type(2)));
typedef float v8f  __attribute__((ext_vector_type(8)));
typedef unsigned int u32x4 __attribute__((ext_vector_type(4)));
typedef int   i32x4v __attribute__((ext_vector_type(4)));
typedef int   i32x8v __attribute__((ext_vector_type(8)));

#define NEG_INF (-__builtin_inff())

#if defined(__has_builtin)
#if __has_builtin(__builtin_amdgcn_wmma_f32_16x16x4_f32)
#define HAVE_WMMA_F32 1
#endif
#if __has_builtin(__builtin_amdgcn_tensor_load_to_lds) && \
    __has_builtin(__builtin_amdgcn_s_wait_tensorcnt)
#define HAVE_TDM 1
#endif
#endif

#if defined(__has_include)
#if __has_include(<hip/amd_detail/amd_gfx1250_TDM.h>)
#define TDM_SIX_ARG 1   // therock-10.0 headers => 6-arg builtin arity
#endif
#endif

// ---------------------------------------------------------------------------
// Kernel 1: Q = x@Wq + bq, K = x@Wk + bk   (trivial FLOPs; coalesced on p)
// ---------------------------------------------------------------------------
__global__ __launch_bounds__(256)
void tka_project(const float* __restrict__ x,
                 const float* __restrict__ Wq, const float* __restrict__ bq,
                 const float* __restrict__ Wk, const float* __restrict__ bk,
                 float* __restrict__ Q, float* __restrict__ K)
{
    const int id    = blockIdx.x * blockDim.x + threadIdx.x;
    const int total = B_ * L_ * P_;
    const bool isK  = id >= total;
    const int  t    = isK ? id - total : id;
    const int  row  = t / P_;
    const int  p    = t % P_;

    const float* __restrict__ W    = isK ? Wk : Wq;
    const float* __restrict__ bias = isK ? bk : bq;
    const float* __restrict__ xr   = x + (size_t)row * D_;

    float acc = bias[p];
    #pragma unroll 8
    for (int d = 0; d < D_; ++d)
        acc = __builtin_fmaf(xr[d], W[d * P_ + p], acc);
    (isK ? K : Q)[t] = acc;
}

// ---------------------------------------------------------------------------
// Block-cooperative exact top-64 extraction from an LDS slab.
// Per-thread cached argmax over a strided slice; each round: wave 0 argmax-
// reduces the 256 cached maxima (8 strided reads + 5 shfl_xor steps, wave32),
// records the winner, and only the owning thread invalidates + rescans.
// Only 2 barriers per extracted element. Ties -> lower index (matches top_k).
// ---------------------------------------------------------------------------
__device__ __forceinline__ void block_topk_extract(
    float* slab, int n,
    float* s_rv, int* s_ri,
    volatile float* s_bv, volatile int* s_bi,
    float* outv, int* outi)
{
    const int tid = threadIdx.x;

    float bv = NEG_INF; int bi = tid < n ? tid : 0;
    for (int e = tid; e < n; e += 256) {
        float v = slab[e];
        if (v > bv) { bv = v; bi = e; }
    }
    s_rv[tid] = bv; s_ri[tid] = bi;
    __syncthreads();

    for (int it = 0; it < TK_; ++it) {
        if (tid < 32) {                       // wave 0: argmax over 256 cached
            float v = s_rv[tid]; int i = s_ri[tid];
            #pragma unroll
            for (int e = 1; e < 8; ++e) {
                float v2 = s_rv[tid + e * 32]; int i2 = s_ri[tid + e * 32];
                if (v2 > v || (v2 == v && i2 < i)) { v = v2; i = i2; }
            }
            #pragma unroll
            for (int off = 16; off > 0; off >>= 1) {
                float v2 = __shfl_xor(v, off, 32);
                int   i2 = __shfl_xor(i, off, 32);
                if (v2 > v || (v2 == v && i2 < i)) { v = v2; i = i2; }
            }
            if (tid == 0) { *s_bv = v; *s_bi = i; outv[it] = v; outi[it] = i; }
        }
        __syncthreads();
        const int wi = *s_bi;
        if (tid == (wi & 255)) {              // owner of the extracted slot
            slab[wi] = NEG_INF;
            float nv = NEG_INF; int ni = tid < n ? tid : 0;
            for (int e = tid; e < n; e += 256) {
                float v = slab[e];
                if (v > nv) { nv = v; ni = e; }
            }
            s_rv[tid] = nv; s_ri[tid] = ni;
        }
        __syncthreads();
    }
}

// ---------------------------------------------------------------------------
// Kernel 2: 16-row score slab via f32 WMMA (16x16x4 chained, K=64); Q tile
// staged to LDS by the Tensor Data Mover; mask+scale in-register; block-local
// exact top-64. grid = B*(L/16), block = 256 (8 waves, 16 col tiles each).
// ---------------------------------------------------------------------------
__global__ __launch_bounds__(256)
void tka_scores_topk(const float* __restrict__ Q, const float* __restrict__ K,
                     const int* __restrict__ pm,
                     float* __restrict__ cand_v, int* __restrict__ cand_i)
{
    __shared__ float s_scores[SLAB];      // 128 KB
    __shared__ float s_qtile[ROWT * P_];  // 4 KB Q tile (TDM destination)
    __shared__ float s_rv[256]; __shared__ int s_ri[256];
    __shared__ int   s_rm16[ROWT];
    __shared__ float s_cv[TK_]; __shared__ int s_ci[TK_];
    __shared__ float s_bv;      __shared__ int s_bi;

    const int tid   = threadIdx.x;
    const int lane  = tid & 31;
    const int wave  = tid >> 5;
    const int batch = blockIdx.x / NBLK_ROW;
    const int rt    = blockIdx.x % NBLK_ROW;
    const int row0  = rt * ROWT;

    const float* __restrict__ Qb  = Q + ((size_t)batch * L_ + row0) * P_;
    const float* __restrict__ Kb  = K + (size_t)batch * L_ * P_;
    const int*   __restrict__ pmb = pm + batch * L_;

    // ---- Stage the 16x64 f32 Q tile into LDS (TDM) + row mask ----
#if defined(HAVE_TDM)
    if (wave == 0) {
        const unsigned long long ga = (unsigned long long)(size_t)Qb;
        const unsigned lds_addr = (unsigned)(size_t)(void*)s_qtile;
        u32x4 g0;
        g0[0] = 1u;                                           // count=1 (user D#)
        g0[1] = lds_addr;                                     // lds_addr
        g0[2] = (unsigned)(ga & 0xffffffffull);               // global_addr lo
        g0[3] = (unsigned)((ga >> 32) & 0x1ffffffull) | (2u << 30); // hi | type=2
        i32x8v g1;
        g1[0] = (int)(2u << 16);        // data_size = 4B
        g1[1] = (int)((unsigned)P_ << 16);   // tensor_dim0 = 64  (bits 79:48)
        g1[2] = (int)((unsigned)L_ << 16);   // tensor_dim1 = 2048 (bits 111:80)
        g1[3] = (int)((unsigned)P_ << 16);   // tile_dim0 = 64   (bits 127:112)
        g1[4] = ROWT;                        // tile_dim1 = 16
        g1[5] = P_;                          // tensor_dim0_stride = 64
        g1[6] = 0; g1[7] = 0;
        i32x4v gz = {0, 0, 0, 0};
#if defined(TDM_SIX_ARG)
        i32x8v gz8 = {0, 0, 0, 0, 0, 0, 0, 0};
        __builtin_amdgcn_tensor_load_to_lds(g0, g1, gz, gz, gz8, 0);
#else
        __builtin_amdgcn_tensor_load_to_lds(g0, g1, gz, gz, 0);
#endif
        __builtin_amdgcn_s_wait_tensorcnt(0);
    }
#else
    if (tid < ROWT * P_ / 4) {            // fallback: plain cooperative copy
        ((float4*)s_qtile)[tid] = ((const float4*)Qb)[tid];
    }
#endif
    if (tid < ROWT) s_rm16[tid] = pmb[row0 + tid];
    __syncthreads();

    const int mn    = lane & 15;          // M for A-lane / N for B,C,D-lane
    const int khalf = (lane >> 4) * 2;    // upper lane half holds K+2,K+3
    const float* qrow = s_qtile + mn * P_;

    bool rOk[8];
    #pragma unroll
    for (int j = 0; j < 8; ++j)
        rOk[j] = s_rm16[j + 8 * (lane >> 4)] != 0;

    for (int ct = wave; ct < NCOLT; ct += 8) {
        const int col0 = ct * 16;
        const float* __restrict__ krow = Kb + (size_t)(col0 + mn) * P_;
        if (ct + 8 < NCOLT)
            __builtin_prefetch(Kb + (size_t)(col0 + 128 + mn) * P_, 0, 1);

        v8f c = {0.f, 0.f, 0.f, 0.f, 0.f, 0.f, 0.f, 0.f};
#if defined(HAVE_WMMA_F32)
        #pragma unroll
        for (int kk = 0; kk < 16; ++kk) {           // K = 64 in steps of 4
            const int k0 = kk * 4 + khalf;
            v2f a; a.x = qrow[k0]; a.y = qrow[k0 + 1];
            v2f b; b.x = krow[k0]; b.y = krow[k0 + 1];
            c = __builtin_amdgcn_wmma_f32_16x16x4_f32(
                    false, a, false, b, (short)0, c, false, false);
        }
#else
        #pragma unroll
        for (int j = 0; j < 8; ++j) {
            const int mm = j + 8 * (lane >> 4);
            const float* qr = s_qtile + mm * P_;
            float acc = 0.f;
            for (int p = 0; p < P_; ++p)
                acc = __builtin_fmaf(qr[p], krow[p], acc);
            c[j] = acc;
        }
#endif
        const bool colOk = pmb[col0 + mn] != 0;
        #pragma unroll
        for (int j = 0; j < 8; ++j) {
            const int mm = j + 8 * (lane >> 4);
            s_scores[mm * L_ + col0 + mn] =
                (colOk && rOk[j]) ? c[j] * 0.125f : NEG_INF;
        }
    }
    __syncthreads();

    block_topk_extract(s_scores, SLAB, s_rv, s_ri, &s_bv, &s_bi, s_cv, s_ci);

    if (tid < TK_) {
        const int o = (batch * NBLK_ROW + rt) * TK_ + tid;
        cand_v[o] = s_cv[tid];
        cand_i[o] = row0 * L_ + s_ci[tid];   // local (m*L+col) -> batch-flat index
    }
}

// ---------------------------------------------------------------------------
// Kernel 3: per-batch merge of 8192 candidates -> global top-64 + softmax.
// Winners come out sorted descending (matches jax.lax.top_k ordering).
// Output: [B*64*2] indices (as float i, j) then [B*64] weights.
// ---------------------------------------------------------------------------
__global__ __launch_bounds__(256)
void tka_merge(const float* __restrict__ cand_v, const int* __restrict__ cand_i,
               float* __restrict__ out)
{
    __shared__ float s_v[NCAND];  __shared__ int s_i[NCAND];   // 64 KB
    __shared__ float s_rv[256];   __shared__ int s_ri[256];
    __shared__ float s_wv[TK_];   __shared__ int s_wi[TK_];
    __shared__ float s_bv;        __shared__ int s_bi;
    __shared__ float s_sum;

    const int tid   = threadIdx.x;
    const int batch = blockIdx.x;

    for (int e = tid; e < NCAND; e += 256) {
        s_v[e] = cand_v[batch * NCAND + e];
        s_i[e] = cand_i[batch * NCAND + e];
    }
    __syncthreads();

    block_topk_extract(s_v, NCAND, s_rv, s_ri, &s_bv, &s_bi, s_wv, s_wi);

    if (tid < TK_) s_wi[tid] = s_i[s_wi[tid]];   // slab slot -> batch-flat index
    __syncthreads();

    if (tid == 0) {
        const float mx = s_wv[0];
        float sum = 0.f;
        for (int k = 0; k < TK_; ++k) sum += __expf(s_wv[k] - mx);
        s_sum = sum;
    }
    __syncthreads();

    if (tid < TK_) {
        const int gi = s_wi[tid];
        out[((size_t)batch * TK_ + tid) * 2 + 0] = (float)(gi / L_);
        out[((size_t)batch * TK_ + tid) * 2 + 1] = (float)(gi % L_);
        out[(size_t)B_ * TK_ * 2 + batch * TK_ + tid] =
            __expf(s_wv[tid] - s_wv[0]) / s_sum;
    }
}

// ---------------------------------------------------------------------------
extern "C" void kernel_launch(void* const* d_in, const int* in_sizes, int n_in,
                              void* d_out, int out_size, void* d_ws, size_t ws_size,
                              hipStream_t stream)
{
    const float* x  = (const float*)d_in[0];
    const int*   pm = (const int*)  d_in[1];
    const float* Wq = (const float*)d_in[2];
    const float* bq = (const float*)d_in[3];
    const float* Wk = (const float*)d_in[4];
    const float* bk = (const float*)d_in[5];
    (void)in_sizes; (void)n_in; (void)out_size; (void)ws_size;

    float* Q  = (float*)d_ws;                       // B*L*P f32 = 4 MB
    float* K  = Q + (size_t)B_ * L_ * P_;           // 4 MB
    float* cv = K + (size_t)B_ * L_ * P_;           // B*8192 f32 = 256 KB
    int*   ci = (int*)(cv + (size_t)B_ * NCAND);    // 256 KB

    const int projThreads = 2 * B_ * L_ * P_;       // Q and K halves
    tka_project<<<projThreads / 256, 256, 0, stream>>>(x, Wq, bq, Wk, bk, Q, K);

    tka_scores_topk<<<B_ * NBLK_ROW, 256, 0, stream>>>(Q, K, pm, cv, ci);

    tka_merge<<<B_, 256, 0, stream>>>(cv, ci, (float*)d_out);
}